// InferenceNet_71459665870944
// MI455X (gfx1250) — compile-verified
//
#include <hip/hip_runtime.h>
#include <hip/hip_bf16.h>
#include <stdint.h>

typedef __attribute__((ext_vector_type(2))) float v2f;
typedef __attribute__((ext_vector_type(4))) float v4fv;
typedef __attribute__((ext_vector_type(8))) float v8f;
typedef __attribute__((ext_vector_type(4))) int   v4i;

#define BB    8
#define TT    2048
#define IDIM  512
#define HDIM  2048
#define ODIM  512
#define ROWS  (BB * TT)              // 16384
#define XS_STRIDE (IDIM + 4)         // 516: %64==4 -> conflict-free A loads; *4 %16==0 -> B128 LDS align
#define HS_STRIDE (HDIM + 4)         // 2052: same properties

__device__ __forceinline__ v8f wmma_f32(v2f a, v2f b, v8f c) {
  // D = A(16x4 f32) * B(4x16 f32) + C(16x16 f32)
  return __builtin_amdgcn_wmma_f32_16x16x4_f32(
      /*neg_a=*/false, a, /*neg_b=*/false, b,
      /*c_mod=*/(short)0, c, /*reuse_a=*/false, /*reuse_b=*/false);
}

__global__ __launch_bounds__(256, 1)
void fused_enc_topk_dec(const float* __restrict__ x,          // [ROWS, IDIM]
                        const int*   __restrict__ mask_prev,  // [ROWS, HDIM]
                        const float* __restrict__ W_enc,      // [HDIM, IDIM]
                        const float* __restrict__ b_enc,      // [HDIM]
                        const float* __restrict__ W_dec,      // [ODIM, HDIM]
                        const float* __restrict__ b_dec,      // [ODIM]
                        float* __restrict__ out,              // [ROWS, ODIM]
                        float* __restrict__ mask_out)         // [ROWS, HDIM]
{
  extern __shared__ float smem[];
  float* xs = smem;                       // [16][XS_STRIDE]
  float* hs = smem + 16 * XS_STRIDE;      // [16][HS_STRIDE]

  const int tid   = threadIdx.x;
  const int lane  = tid & 31;
  const int wave  = tid >> 5;
  const int row0  = blockIdx.x * 16;

  // ---------- Phase 0: async-copy x tile straight into LDS (CDNA5 path) ----------
  // LDS aperture: flat addr[31:0] == LDS byte offset (ISA 10.2), so truncation is valid.
  {
    const unsigned lds_base = (unsigned)(uintptr_t)xs;
    for (int i = tid; i < 16 * (IDIM / 4); i += 256) {
      const int r = i >> 7;               // IDIM/4 == 128
      const int c = (i & 127) * 4;
      const uint64_t gaddr = (uint64_t)(uintptr_t)(x + (size_t)(row0 + r) * IDIM + c);
      const unsigned laddr = lds_base + (unsigned)((r * XS_STRIDE + c) * 4);
      asm volatile("global_load_async_to_lds_b128 %0, %1, off"
                   :: "v"(laddr), "v"(gaddr) : "memory");
    }
    asm volatile("s_wait_asynccnt 0x0" ::: "memory");
  }
  __syncthreads();

  // WMMA fragment addressing (per ISA VGPR layouts, wave32):
  //  A 16x4 f32:  lane&15 = M row, (lane>>4)*2 + {0,1} = K   -> b64 per lane
  //  B 4x16 f32:  lane&15 = N col, (lane>>4)*2 + {0,1} = K   -> b64 per lane
  //  D 16x16 f32: vgpr r, lanes 0-15 -> M=r, lanes 16-31 -> M=r+8; N=lane&15
  const int an    = lane & 15;
  const int koff  = (lane >> 4) * 2;
  const int mbase = (lane >> 4) * 8;

  // ---------- Phase 1: h = x @ W_enc^T + b_enc (mask_prev applied later) ----------
  for (int ct = wave; ct < HDIM / 16; ct += 8) {
    const int n0 = ct * 16;
    const float* wb = W_enc + (size_t)(n0 + an) * IDIM + koff;  // B[k][n] = W_enc[n][k]
    const float* ab = xs + an * XS_STRIDE + koff;
    __builtin_prefetch(wb, 0, 1);
    v8f acc = {};
    #pragma unroll 8
    for (int kb = 0; kb < IDIM; kb += 4) {
      v2f a = *reinterpret_cast<const v2f*>(ab + kb);
      v2f b = *reinterpret_cast<const v2f*>(wb + kb);
      acc = wmma_f32(a, b, acc);
    }
    const int   col  = n0 + an;
    const float bias = b_enc[col];
    #pragma unroll
    for (int r = 0; r < 8; ++r) {
      hs[(mbase + r) * HS_STRIDE + col] = acc[r] + bias;
    }
  }
  __syncthreads();

  // ---------- Phase 2: mask_prev zeroing + exact top-512/top-256 of h^2 ----------
  // Lane owns 4 contiguous elements per chunk -> B128 global/LDS traffic.
  // mask_prev/mask_out are single-touch streams: NT hints keep W_enc/W_dec in L2.
  for (int m = wave; m < 16; m += 8) {
    float* hr = hs + m * HS_STRIDE;
    const size_t gm = (size_t)(row0 + m);
    unsigned ev[64];                 // energy bit patterns (monotone for >= 0)
    uint64_t mpb = 0;                // packed mask_prev flags, bit (q*4+j)
    #pragma unroll
    for (int q = 0; q < 16; ++q) {
      const int c0 = q * 128 + lane * 4;
      const v4i  mp4 = __builtin_nontemporal_load(
          reinterpret_cast<const v4i*>(mask_prev + gm * HDIM + c0));
      const v4fv v4 = *reinterpret_cast<const v4fv*>(hr + c0);  // ds_load_b128
      #pragma unroll
      for (int j = 0; j < 4; ++j) {
        float v = v4[j];
        if (mp4[j] > 0) { v = 0.0f; mpb |= (1ull << (q * 4 + j)); }
        ev[q * 4 + j] = __float_as_uint(v * v);
      }
    }
    // bitwise binary search: max thr with count(e >= thr) >= k  == k-th largest
    unsigned thr[2];
    const int kk[2] = {512, 256};
    for (int s = 0; s < 2; ++s) {
      unsigned t = 0;
      for (int bit = 30; bit >= 0; --bit) {
        const unsigned cand = t | (1u << bit);
        int cnt = 0;
        #pragma unroll
        for (int ch = 0; ch < 64; ++ch)
          cnt += __popcll(__ballot(ev[ch] >= cand));
        if (cnt >= kk[s]) t = cand;        // cnt is wave-uniform
      }
      thr[s] = t;
    }
    #pragma unroll
    for (int q = 0; q < 16; ++q) {
      const int c0 = q * 128 + lane * 4;
      const v4fv v4 = *reinterpret_cast<const v4fv*>(hr + c0);
      v4fv hv, mo;
      #pragma unroll
      for (int j = 0; j < 4; ++j) {
        const int ch = q * 4 + j;
        const float mpf = (float)((mpb >> ch) & 1ull);
        const float v   = (mpf != 0.0f) ? 0.0f : v4[j];   // h after mask_prev zeroing
        const unsigned e = ev[ch];
        hv[j] = (e >= thr[0]) ? v : 0.0f;                 // h *= mask_share
        mo[j] = mpf + ((e >= thr[1]) ? 1.0f : 0.0f);      // mask_prev_new
      }
      *reinterpret_cast<v4fv*>(hr + c0) = hv;             // ds_store_b128
      __builtin_nontemporal_store(mo,
          reinterpret_cast<v4fv*>(mask_out + gm * HDIM + c0));
    }
  }
  __syncthreads();

  // ---------- Phase 3: out = h @ W_dec^T + b_dec ----------
  for (int ct = wave; ct < ODIM / 16; ct += 8) {
    const int n0 = ct * 16;
    const float* wb = W_dec + (size_t)(n0 + an) * HDIM + koff;  // B[k][n] = W_dec[n][k]
    const float* ab = hs + an * HS_STRIDE + koff;
    __builtin_prefetch(wb, 0, 1);
    v8f acc = {};
    #pragma unroll 8
    for (int kb = 0; kb < HDIM; kb += 4) {
      v2f a = *reinterpret_cast<const v2f*>(ab + kb);
      v2f b = *reinterpret_cast<const v2f*>(wb + kb);
      acc = wmma_f32(a, b, acc);
    }
    const float bias = b_dec[n0 + an];
    #pragma unroll
    for (int r = 0; r < 8; ++r) {
      __builtin_nontemporal_store(acc[r] + bias,
          out + (size_t)(row0 + mbase + r) * ODIM + n0 + an);
    }
  }
}

extern "C" void kernel_launch(void* const* d_in, const int* in_sizes, int n_in,
                              void* d_out, int out_size, void* d_ws, size_t ws_size,
                              hipStream_t stream) {
  (void)in_sizes; (void)n_in; (void)out_size; (void)d_ws; (void)ws_size;
  const float* x         = (const float*)d_in[0];
  const int*   mask_prev = (const int*)  d_in[1];
  const float* W_enc     = (const float*)d_in[2];
  const float* b_enc     = (const float*)d_in[3];
  const float* W_dec     = (const float*)d_in[4];
  const float* b_dec     = (const float*)d_in[5];

  float* out      = (float*)d_out;                       // [ROWS, ODIM]
  float* mask_out = out + (size_t)ROWS * ODIM;           // [ROWS, HDIM]

  const size_t lds_bytes = (size_t)(16 * XS_STRIDE + 16 * HS_STRIDE) * sizeof(float); // ~160 KB (CDNA5 WGP LDS)
  fused_enc_topk_dec<<<ROWS / 16, 256, lds_bytes, stream>>>(
      x, mask_prev, W_enc, b_enc, W_dec, b_dec, out, mask_out);
}